// optimNet_7026566496512
// MI455X (gfx1250) — compile-verified
//
#include <hip/hip_runtime.h>

typedef __attribute__((ext_vector_type(2))) float v2f;
typedef __attribute__((ext_vector_type(8))) float v8f;

#define NN 50000
#define NE 800000
#define FIN 1536
#define FH  512
#define FOUT 128

__device__ __forceinline__ void atomAddF(float* p, float v) {
#if defined(__has_builtin) && __has_builtin(__hip_atomic_fetch_add)
  __hip_atomic_fetch_add(p, v, __ATOMIC_RELAXED, __HIP_MEMORY_SCOPE_AGENT);
#else
  atomicAdd(p, v);
#endif
}

// ---------------------------------------------------------------------------
// f32 WMMA GEMM: OUT[M,N] = X[M,K] @ W[K,N], LDS-staged B.
//   Block: WAVES waves; block tile = (WAVES*16) M x 128 N.
//   Wave tile: 16 M x 128 N = 8 C fragments; K-step 4.
//   B staged per 64-K chunk in LDS as (W[k][c],W[k+1][c]) float2 pairs,
//   column XOR-16 swizzle on odd pairs -> ds_load_b64 is conflict-free.
// A-frag (16x4 f32): lane L holds M=L%16, K = {2*(L/16), +1}.
// B-frag (4x16):     lane L holds N=L%16, K = {2*(L/16), +1}.
// C/D:               lane L, vgpr r -> [M = r + 8*(L/16)][N = L%16].
// ---------------------------------------------------------------------------
template <int K, int N, int WAVES>
__global__ __launch_bounds__(WAVES * 32) void gemm_f32_wmma(const float* __restrict__ X,
                                                            const float* __restrict__ W,
                                                            float* __restrict__ OUT) {
  constexpr int KC = 64;    // K chunk
  constexpr int NB = 128;   // N block (= wave N tile)
  __shared__ float lds[(KC / 2) * NB * 2];  // 32 KB

  const int tid   = threadIdx.x;
  const int lane  = tid & 31;
  const int wave  = tid >> 5;
  const int m0    = blockIdx.x * (WAVES * 16) + (wave << 4);
  const int n0    = blockIdx.y * NB;
  const int t16   = lane & 15;
  const int khb   = lane >> 4;           // 0/1: which K-pair half this lane holds
  const int kh    = khb << 1;            // 0 or 2

  v8f c0 = {}, c1 = {}, c2 = {}, c3 = {}, c4 = {}, c5 = {}, c6 = {}, c7 = {};
  const float* xrow = X + (size_t)(m0 + t16) * K + kh;

  for (int kc = 0; kc < K; kc += KC) {
    __syncthreads();
    // ---- stage B chunk: pairs p=0..KC/2-1, cols c=0..NB-1 (coalesced) ----
    for (int i = tid; i < (KC / 2) * NB; i += WAVES * 32) {
      int p = i >> 7;          // pair index
      int cc = i & (NB - 1);   // column within block
      v2f f;
      f.x = W[(size_t)(kc + 2 * p) * N + n0 + cc];
      f.y = W[(size_t)(kc + 2 * p + 1) * N + n0 + cc];
      int csw = cc ^ ((p & 1) << 4);  // swizzle
      *(v2f*)&lds[((p * NB) + csw) << 1] = f;
    }
    __syncthreads();

    // ---- compute chunk ----
    for (int ks = 0; ks < KC; ks += 4) {
      v2f a = *(const v2f*)(xrow + kc + ks);                 // A frag (b64)
      const int p = (ks >> 1) + khb;                         // this lane's pair
      const int sw = p & 1;
      const float* base = &lds[((p * NB) + t16) << 1];
      const float* be = base + (sw << 5);        // even tiles (t^sw==t)
      const float* bo = base + ((1 ^ sw) << 5);  // odd tiles
      v2f b;
      b = *(const v2f*)(be + 0 * 64);
      c0 = __builtin_amdgcn_wmma_f32_16x16x4_f32(false, a, false, b, (short)0, c0, false, false);
      b = *(const v2f*)(bo + 0 * 64);
      c1 = __builtin_amdgcn_wmma_f32_16x16x4_f32(false, a, false, b, (short)0, c1, false, false);
      b = *(const v2f*)(be + 1 * 64);
      c2 = __builtin_amdgcn_wmma_f32_16x16x4_f32(false, a, false, b, (short)0, c2, false, false);
      b = *(const v2f*)(bo + 1 * 64);
      c3 = __builtin_amdgcn_wmma_f32_16x16x4_f32(false, a, false, b, (short)0, c3, false, false);
      b = *(const v2f*)(be + 2 * 64);
      c4 = __builtin_amdgcn_wmma_f32_16x16x4_f32(false, a, false, b, (short)0, c4, false, false);
      b = *(const v2f*)(bo + 2 * 64);
      c5 = __builtin_amdgcn_wmma_f32_16x16x4_f32(false, a, false, b, (short)0, c5, false, false);
      b = *(const v2f*)(be + 3 * 64);
      c6 = __builtin_amdgcn_wmma_f32_16x16x4_f32(false, a, false, b, (short)0, c6, false, false);
      b = *(const v2f*)(bo + 3 * 64);
      c7 = __builtin_amdgcn_wmma_f32_16x16x4_f32(false, a, false, b, (short)0, c7, false, false);
    }
  }

  float* orow = OUT + (size_t)(m0 + (khb << 3)) * N + n0 + t16;
#pragma unroll
  for (int r = 0; r < 8; ++r) {
    float* o = orow + (size_t)r * N;
    o[0]   = c0[r];
    o[16]  = c1[r];
    o[32]  = c2[r];
    o[48]  = c3[r];
    o[64]  = c4[r];
    o[80]  = c5[r];
    o[96]  = c6[r];
    o[112] = c7[r];
  }
}

// ---------------------------------------------------------------------------
// Elementwise / scatter helpers
// ---------------------------------------------------------------------------
__global__ void fill_ones(float* __restrict__ p, int n) {
  int i = blockIdx.x * blockDim.x + threadIdx.x;
  if (i < n) p[i] = 1.0f;
}

__global__ void deg_accum(const long long* __restrict__ col, const float* __restrict__ w,
                          float* __restrict__ deg, int e) {
  int i = blockIdx.x * blockDim.x + threadIdx.x;
  if (i < e) atomAddF(&deg[(int)col[i]], w[i]);
}

__global__ void rsqrt_inplace(float* __restrict__ d, int n) {
  int i = blockIdx.x * blockDim.x + threadIdx.x;
  if (i < n) {
    float v = d[i];
    d[i] = (v > 0.0f) ? rsqrtf(v) : 0.0f;
  }
}

template <int F>
__global__ void self_scale(const float* __restrict__ xw, const float* __restrict__ dinv,
                           float* __restrict__ h, int total) {
  int i = blockIdx.x * blockDim.x + threadIdx.x;
  if (i < total) {
    float di = dinv[i / F];
    h[i] = xw[i] * di * di;
  }
}

template <int F>
__global__ void self_scale_bias(const float* __restrict__ xw, const float* __restrict__ dinv,
                                const float* __restrict__ b, float* __restrict__ o, int total) {
  int i = blockIdx.x * blockDim.x + threadIdx.x;
  if (i < total) {
    float di = dinv[i / F];
    o[i] = xw[i] * di * di + b[i & (F - 1)];
  }
}

template <int F>
__global__ void bias_relu(float* __restrict__ h, const float* __restrict__ b, int total) {
  int i = blockIdx.x * blockDim.x + threadIdx.x;
  if (i < total) h[i] = fmaxf(h[i] + b[i & (F - 1)], 0.0f);
}

// One wave per edge: h[col] += xw[row] * (dinv[row]*w*dinv[col])
template <int F>
__global__ void scatter_edges(const long long* __restrict__ row, const long long* __restrict__ col,
                              const float* __restrict__ w, const float* __restrict__ dinv,
                              const float* __restrict__ xw, float* __restrict__ h, int e) {
  int wid  = blockIdx.x * (blockDim.x >> 5) + (threadIdx.x >> 5);
  int lane = threadIdx.x & 31;
  if (wid >= e) return;
  int r = (int)row[wid], c = (int)col[wid];
  float coeff = dinv[r] * w[wid] * dinv[c];
  const float* src = xw + (size_t)r * F;
  float* dst = h + (size_t)c * F;
#pragma unroll
  for (int f = lane; f < F; f += 32) atomAddF(&dst[f], src[f] * coeff);
}

// One wave per node: sp[n] = h[n]·Wm[0:FH], dp[n] = h[n]·Wm[FH:2FH]
__global__ void proj_kernel(const float* __restrict__ h, const float* __restrict__ wm,
                            float* __restrict__ sp, float* __restrict__ dp, int n) {
  int wid  = blockIdx.x * (blockDim.x >> 5) + (threadIdx.x >> 5);
  int lane = threadIdx.x & 31;
  if (wid >= n) return;
  const float* hr = h + (size_t)wid * FH;
  float s = 0.0f, d = 0.0f;
#pragma unroll 4
  for (int f = lane; f < FH; f += 32) {
    float v = hr[f];
    s += v * wm[f];
    d += v * wm[FH + f];
  }
#pragma unroll
  for (int off = 16; off > 0; off >>= 1) {
    s += __shfl_xor(s, off, 32);
    d += __shfl_xor(d, off, 32);
  }
  if (lane == 0) { sp[wid] = s; dp[wid] = d; }
}

__global__ void edge_weight_kernel(const long long* __restrict__ row, const long long* __restrict__ col,
                                   const float* __restrict__ sp, const float* __restrict__ dp,
                                   const float* __restrict__ bm, float* __restrict__ ew,
                                   float* __restrict__ deg2, int e) {
  int i = blockIdx.x * blockDim.x + threadIdx.x;
  if (i < e) {
    int r = (int)row[i], c = (int)col[i];
    float v = fmaxf(sp[r] + dp[c] + bm[0], 0.0f);
    ew[i] = v;
    atomAddF(&deg2[c], v);
  }
}

// ---------------------------------------------------------------------------
extern "C" void kernel_launch(void* const* d_in, const int* in_sizes, int n_in,
                              void* d_out, int out_size, void* d_ws, size_t ws_size,
                              hipStream_t stream) {
  (void)in_sizes; (void)n_in; (void)out_size; (void)ws_size;
  const float*     x   = (const float*)d_in[0];
  const float*     ea  = (const float*)d_in[1];
  const long long* ei  = (const long long*)d_in[2];   // int64 [2, NE]
  const float*     W1  = (const float*)d_in[3];
  const float*     b1  = (const float*)d_in[4];
  const float*     W2  = (const float*)d_in[5];
  const float*     b2  = (const float*)d_in[6];
  const float*     Wm  = (const float*)d_in[7];
  const float*     bm  = (const float*)d_in[8];
  float* out = (float*)d_out;
  const long long* row = ei;
  const long long* col = ei + NE;

  // workspace layout (floats); xw2 aliases xw1 (dead after conv1 scatter)
  float* ws    = (float*)d_ws;
  float* xw1   = ws;                               // NN*FH
  float* h1    = xw1 + (size_t)NN * FH;            // NN*FH
  float* dinv1 = h1 + (size_t)NN * FH;             // NN
  float* dinv2 = dinv1 + NN;                       // NN
  float* sp    = dinv2 + NN;                       // NN
  float* dp    = sp + NN;                          // NN
  float* ew    = dp + NN;                          // NE
  float* xw2   = xw1;                              // NN*FOUT (alias)

  const int TB = 256;
  const int nb_nodes = (NN + TB - 1) / TB;
  const int nb_edges = (NE + TB - 1) / TB;
  const int nb_h1    = (NN * FH + TB - 1) / TB;
  const int nb_o     = (NN * FOUT + TB - 1) / TB;
  const int nb_ewave = NE / 8;    // 8 waves/block, 1 edge/wave (exact)
  const int nb_nwave = NN / 8;    // 1 node/wave (exact)

  // ---- conv1 normalization ----
  fill_ones<<<nb_nodes, TB, 0, stream>>>(dinv1, NN);
  deg_accum<<<nb_edges, TB, 0, stream>>>(col, ea, dinv1, NE);
  rsqrt_inplace<<<nb_nodes, TB, 0, stream>>>(dinv1, NN);

  // ---- GEMM1: xw1 = x @ W1  (50000x1536 @ 1536x512), WMMA f32, LDS-staged B ----
  // 625 M-blocks of 80 rows (5 waves), 4 N-blocks of 128 cols
  gemm_f32_wmma<FIN, FH, 5><<<dim3(NN / 80, FH / 128), 160, 0, stream>>>(x, W1, xw1);

  // ---- conv1: self loop + edge scatter + bias/relu ----
  self_scale<FH><<<nb_h1, TB, 0, stream>>>(xw1, dinv1, h1, NN * FH);
  scatter_edges<FH><<<nb_ewave, TB, 0, stream>>>(row, col, ea, dinv1, xw1, h1, NE);
  bias_relu<FH><<<nb_h1, TB, 0, stream>>>(h1, b1, NN * FH);

  // ---- edge MLP: per-node projections, per-edge relu weight + deg2 ----
  proj_kernel<<<nb_nwave, TB, 0, stream>>>(h1, Wm, sp, dp, NN);
  fill_ones<<<nb_nodes, TB, 0, stream>>>(dinv2, NN);
  edge_weight_kernel<<<nb_edges, TB, 0, stream>>>(row, col, sp, dp, bm, ew, dinv2, NE);
  rsqrt_inplace<<<nb_nodes, TB, 0, stream>>>(dinv2, NN);

  // ---- GEMM2: xw2 = h1 @ W2  (50000x512 @ 512x128), WMMA f32 ----
  gemm_f32_wmma<FH, FOUT, 5><<<dim3(NN / 80, 1), 160, 0, stream>>>(h1, W2, xw2);

  // ---- conv2: self loop + bias init, then edge scatter into d_out ----
  self_scale_bias<FOUT><<<nb_o, TB, 0, stream>>>(xw2, dinv2, b2, out, NN * FOUT);
  scatter_edges<FOUT><<<nb_ewave, TB, 0, stream>>>(row, col, ew, dinv2, xw2, out, NE);
}